// rosa_4bit_layer_80358838108516
// MI455X (gfx1250) — compile-verified
//
#include <hip/hip_runtime.h>

typedef __attribute__((ext_vector_type(2))) float v2f;
typedef __attribute__((ext_vector_type(8))) float v8f;

#define T_SEQ   4096
#define C_DIM   512
#define H_HEADS 128
#define CHUNK   16
#define NCHUNK  (T_SEQ / CHUNK)

__global__ __launch_bounds__(32) void rosa_sufa_kernel(
    const float* __restrict__ xq, const float* __restrict__ xk,
    const float* __restrict__ xv, const float* __restrict__ emb,
    float* __restrict__ out)
{
    const int lane = threadIdx.x;        // 0..31 (wave32)
    const int bh   = blockIdx.x;         // 0 .. B*H-1
    const int b    = bh >> 7;            // / 128 heads
    const int h    = bh & (H_HEADS - 1);
    const size_t base = (size_t)b * T_SEQ * C_DIM + (size_t)(h * 4);

    __shared__ __align__(16) float Qlds[16][4];   // Q rows (t, j)
    __shared__ __align__(16) float Kt[4][16];     // K transposed (j, s)
    __shared__ __align__(16) float Vt[4][16];     // V transposed (e, s)
    __shared__ __align__(16) float Alds[16 * 20]; // masked scores, row stride 20 (pad)
    __shared__ __align__(16) float Slds[16];      // S_prev broadcast, [j*4+e]

    const int m  = lane & 15;            // row (A-op) / col (B-op) / t (epilogue)
    const int hi = lane >> 4;            // half-wave id
    const int k0 = hi * 2;               // K-offset for f32 WMMA operands
    const int ep = hi;                   // output element pair
    const int e0 = 2 * ep;

    const float emb0 = emb[h * 4 + e0];
    const float emb1 = emb[h * 4 + e0 + 1];

    // Running 4x4 state: lane<16 owns S[j][e], j=lane>>2, e=lane&3 (exact f32)
    float Sreg = 0.0f;

    float4 qrow, krow, vrow;
    if (lane < 16) {
        size_t roff = base + (size_t)lane * C_DIM;
        qrow = *(const float4*)(xq + roff);
        krow = *(const float4*)(xk + roff);
        vrow = *(const float4*)(xv + roff);
    }

    for (int c = 0; c < NCHUNK; ++c) {
        // ---- stage chunk + previous state into LDS ----
        if (lane < 16) {
            *(float4*)(&Qlds[lane][0]) = qrow;
            Kt[0][lane] = krow.x; Kt[1][lane] = krow.y;
            Kt[2][lane] = krow.z; Kt[3][lane] = krow.w;
            Vt[0][lane] = vrow.x; Vt[1][lane] = vrow.y;
            Vt[2][lane] = vrow.z; Vt[3][lane] = vrow.w;
            Slds[lane]  = Sreg;
        }
        __syncthreads();

        // ---- prefetch next chunk (hide HBM latency behind compute) ----
        if (c + 1 < NCHUNK && lane < 16) {
            size_t roff = base + (size_t)((c + 1) * CHUNK + lane) * C_DIM;
            qrow = *(const float4*)(xq + roff);
            krow = *(const float4*)(xk + roff);
            vrow = *(const float4*)(xv + roff);
        }

        // ---- scores: A(16x16) = Q(16x4) x K^T(4x16), exact f32 WMMA ----
        v2f aop, bop;
        aop.x = Qlds[m][k0];     aop.y = Qlds[m][k0 + 1];
        bop.x = Kt[k0][m];       bop.y = Kt[k0 + 1][m];
        v8f acc = {};
        acc = __builtin_amdgcn_wmma_f32_16x16x4_f32(
                  false, aop, false, bop, (short)0, acc, false, false);

        // ---- causal mask (keep s <= t) in C/D layout; spill rows to LDS ----
#pragma unroll
        for (int r = 0; r < 8; ++r) {
            int M = r + 8 * hi;                  // query index t
            float a = (m <= M) ? acc[r] : 0.0f;  // key index s = m
            Alds[M * 20 + m] = a;
        }
        __syncthreads();

        // ---- epilogue: o[t][e0], o[t][e0+1] for t = m ----
        float o0 = 0.0f, o1 = 0.0f;
        // cross-chunk term with S_prev
#pragma unroll
        for (int j = 0; j < 4; ++j) {
            float qv = Qlds[m][j];
            o0 += qv * Slds[j * 4 + e0];
            o1 += qv * Slds[j * 4 + e0 + 1];
        }
        // intra-chunk term: masked A row (t) dot V columns
        const float4* Arow = (const float4*)(Alds + m * 20);
        const float4* Va   = (const float4*)(&Vt[e0][0]);
        const float4* Vb   = (const float4*)(&Vt[e0 + 1][0]);
#pragma unroll
        for (int i = 0; i < 4; ++i) {
            float4 a4 = Arow[i];
            float4 va = Va[i];
            float4 vb = Vb[i];
            o0 += a4.x * va.x + a4.y * va.y + a4.z * va.z + a4.w * va.w;
            o1 += a4.x * vb.x + a4.y * vb.y + a4.z * vb.z + a4.w * vb.w;
        }

        // ---- state update: S[j][e] += sum_s K[s][j] * V[s][e] ----
        {
            int j = (lane >> 2) & 3;   // valid for lane<16; lanes>=16 do harmless dup
            int e = lane & 3;
            const float4* Kj = (const float4*)(&Kt[j][0]);
            const float4* Ve = (const float4*)(&Vt[e][0]);
            float accS = 0.0f;
#pragma unroll
            for (int i = 0; i < 4; ++i) {
                float4 kk = Kj[i];
                float4 vv = Ve[i];
                accS += kk.x * vv.x + kk.y * vv.y + kk.z * vv.z + kk.w * vv.w;
            }
            Sreg += accS;
        }

        // ---- write output (x emb) ----
        {
            size_t ooff = base + (size_t)(c * CHUNK + m) * C_DIM + e0;
            float2 ov;
            ov.x = o0 * emb0;
            ov.y = o1 * emb1;
            *(float2*)(out + ooff) = ov;
        }
        __syncthreads();  // protect LDS before next iteration's staging stores
    }
}

extern "C" void kernel_launch(void* const* d_in, const int* in_sizes, int n_in,
                              void* d_out, int out_size, void* d_ws, size_t ws_size,
                              hipStream_t stream) {
    (void)n_in; (void)out_size; (void)d_ws; (void)ws_size;
    const float* xq  = (const float*)d_in[0];
    const float* xk  = (const float*)d_in[1];
    const float* xv  = (const float*)d_in[2];
    const float* emb = (const float*)d_in[3];
    float* out = (float*)d_out;

    const int B = in_sizes[0] / (T_SEQ * C_DIM);   // = 4
    dim3 grid(B * H_HEADS);                        // one wave per (b, head)
    dim3 block(32);
    hipLaunchKernelGGL(rosa_sufa_kernel, grid, block, 0, stream,
                       xq, xk, xv, emb, out);
}